// cal_sim_temp_48344151884380
// MI455X (gfx1250) — compile-verified
//
#include <hip/hip_runtime.h>
#include <hip/hip_bf16.h>

#define INNER 128
#define GRU   256
#define HAN   512
#define BATCH 32
#define NTOK  8192

typedef float v2f __attribute__((ext_vector_type(2)));
typedef float v8f __attribute__((ext_vector_type(8)));

// ---------------------------------------------------------------------------
// Kernel A: M = Wq^T @ Wk   ([128,256]^T x [128,512] -> [256,512])
// One wave per 16x16 tile of M, K = 128 stepped by 4 (v_wmma_f32_16x16x4_f32).
// A-matrix 16x4 layout (ISA 7.12.2): lanes 0-15 hold K=kb+0/kb+1 in v0/v1,
// lanes 16-31 hold K=kb+2/kb+3. B mirrored (row striped across lanes).
// ---------------------------------------------------------------------------
__global__ __launch_bounds__(32)
void wqwk_kernel(const float* __restrict__ Wq, const float* __restrict__ Wk,
                 float* __restrict__ M) {
  const int lane = threadIdx.x;
  const int tg   = blockIdx.x >> 5;   // 0..15 : row tile of M (256/16)
  const int th   = blockIdx.x & 31;   // 0..31 : col tile of M (512/16)
  const int g0 = tg * 16, h0 = th * 16;
  const int mIdx = lane & 15;
  const int kOff = (lane < 16) ? 0 : 2;

  v8f c = {};
  for (int kb = 0; kb < INNER; kb += 4) {
    v2f a, b;
    // A[m=g, k=i] = Wq[i, g]   (Wq is [128,256] row-major)
    a.x = Wq[(kb + kOff)     * GRU + g0 + mIdx];
    a.y = Wq[(kb + kOff + 1) * GRU + g0 + mIdx];
    // B[k=i, n=h] = Wk[i, h]   (Wk is [128,512] row-major)
    b.x = Wk[(kb + kOff)     * HAN + h0 + mIdx];
    b.y = Wk[(kb + kOff + 1) * HAN + h0 + mIdx];
    c = __builtin_amdgcn_wmma_f32_16x16x4_f32(false, a, false, b,
                                              (short)0, c, false, false);
  }
  // C/D layout: VGPR vi -> row vi (lanes 0-15) / row vi+8 (lanes 16-31)
  const int rowAdd = (lane < 16) ? 0 : 8;
  #pragma unroll
  for (int vi = 0; vi < 8; ++vi)
    M[(g0 + vi + rowAdd) * HAN + h0 + mIdx] = c[vi];
}

// ---------------------------------------------------------------------------
// Kernel B: V[n,g] = sum_h edges[n,h] * M[g,h]   ([8192,512] x [256,512]^T)
// One wave per 16(n) x 16(g) tile, K = 512 stepped by 4. M (512 KB) and
// edges (16.8 MB) both stay resident in the 192 MB L2 across tile re-reads.
// ---------------------------------------------------------------------------
__global__ __launch_bounds__(32)
void proj_kernel(const float* __restrict__ E, const float* __restrict__ M,
                 float* __restrict__ V) {
  const int lane = threadIdx.x;
  const int tn   = blockIdx.x & 511;  // 8192/16 n-tiles
  const int tg   = blockIdx.x >> 9;   // 16 g-tiles
  const int n0 = tn * 16, g0 = tg * 16;
  const int mIdx = lane & 15;
  const int kOff = (lane < 16) ? 0 : 2;

  const float* arow = E + (size_t)(n0 + mIdx) * HAN + kOff;  // A[m=n, k=h]
  const float* brow = M + (size_t)(g0 + mIdx) * HAN + kOff;  // B[k=h, n=g]=M[g,h]

  v8f c = {};
  for (int kb = 0; kb < HAN; kb += 4) {
    v2f a, b;
    a.x = arow[kb]; a.y = arow[kb + 1];
    b.x = brow[kb]; b.y = brow[kb + 1];
    c = __builtin_amdgcn_wmma_f32_16x16x4_f32(false, a, false, b,
                                              (short)0, c, false, false);
  }
  const int rowAdd = (lane < 16) ? 0 : 8;
  #pragma unroll
  for (int vi = 0; vi < 8; ++vi)
    V[(size_t)(n0 + vi + rowAdd) * GRU + g0 + mIdx] = c[vi];
}

// ---------------------------------------------------------------------------
// Kernel C (HBM-bound stream, sets the 12 us floor): one wave per token n,
// lane == batch index b (wave32!). Each lane dots change[b,n,:] (256 f32,
// float4 loads) with V[n,:] (L2/L0-resident broadcast), then the masked
// softmax over the 32 batch entries is done with in-wave xor-shuffle
// reductions — no LDS, no atomics. Output [N,B] stores are fully coalesced.
// ---------------------------------------------------------------------------
__global__ __launch_bounds__(256)
void sim_softmax_kernel(const float* __restrict__ change,
                        const int* __restrict__ mask,
                        const float* __restrict__ V,
                        float* __restrict__ out) {
  const int lane = threadIdx.x & 31;          // batch index b
  const int wave = threadIdx.x >> 5;          // 0..7
  const int n    = blockIdx.x * 8 + wave;

  const float4* cp = (const float4*)(change + ((size_t)lane * NTOK + n) * GRU);
  const float4* vp = (const float4*)(V + (size_t)n * GRU);

  float acc = 0.f;
  #pragma unroll 8
  for (int i = 0; i < GRU / 4; ++i) {
    float4 c4 = cp[i];
    float4 v4 = vp[i];
    acc = fmaf(c4.x, v4.x, acc);
    acc = fmaf(c4.y, v4.y, acc);
    acc = fmaf(c4.z, v4.z, acc);
    acc = fmaf(c4.w, v4.w, acc);
  }

  const float scale = 0.088388347648318447f;  // 128^-0.5
  const bool mk = mask[(size_t)lane * NTOK + n] != 0;
  float x = mk ? acc * scale : -1e9f;

  // softmax across the 32 lanes (= batch dim)
  float mx = x;
  #pragma unroll
  for (int off = 16; off > 0; off >>= 1)
    mx = fmaxf(mx, __shfl_xor(mx, off, 32));
  float e = __expf(x - mx);
  float s = e;
  #pragma unroll
  for (int off = 16; off > 0; off >>= 1)
    s += __shfl_xor(s, off, 32);

  out[(size_t)n * BATCH + lane] = e / s;      // out is [N, B] row-major
}

// ---------------------------------------------------------------------------
extern "C" void kernel_launch(void* const* d_in, const int* in_sizes, int n_in,
                              void* d_out, int out_size, void* d_ws, size_t ws_size,
                              hipStream_t stream) {
  const float* change = (const float*)d_in[0];  // [32, 8192, 256]
  const float* edges  = (const float*)d_in[1];  // [8192, 512]
  const int*   mask   = (const int*)d_in[2];    // [32, 8192]
  const float* Wq     = (const float*)d_in[3];  // [128, 256]
  const float* Wk     = (const float*)d_in[4];  // [128, 512]

  float* M = (float*)d_ws;                      // 256*512 f32   (512 KB)
  float* V = (float*)d_ws + (size_t)GRU * HAN;  // 8192*256 f32  (8 MB)

  wqwk_kernel<<<512, 32, 0, stream>>>(Wq, Wk, M);                       // M = Wq^T Wk
  proj_kernel<<<NTOK / 16 * (GRU / 16), 32, 0, stream>>>(edges, M, V);  // V = edges M^T
  sim_softmax_kernel<<<NTOK / 8, 256, 0, stream>>>(change, mask, V, (float*)d_out);
}